// DecoderAttentionRNN_74466142978609
// MI455X (gfx1250) — compile-verified
//
#include <hip/hip_runtime.h>
#include <hip/hip_bf16.h>
#include <math.h>

// MI455X / gfx1250, wave32. fp32 WMMA path: V_WMMA_F32_16X16X4_F32.
typedef float v2f __attribute__((ext_vector_type(2)));
typedef float v8f __attribute__((ext_vector_type(8)));
typedef unsigned int v4u __attribute__((ext_vector_type(4)));
typedef int v4i __attribute__((ext_vector_type(4)));
typedef int v8i __attribute__((ext_vector_type(8)));

#define VOCAB 10000
#define H 256
#define S 4096
#define LDSP 260   // padded row stride (floats) to break LDS bank conflicts

// ---------------------------------------------------------------------------
// wa[k] = dot(Wa_w[k,:], hidden) + Wa_b[k] + Ua_b[k]
__global__ __launch_bounds__(H) void prep_wa_kernel(
    const float* __restrict__ hid, const float* __restrict__ Wa_w,
    const float* __restrict__ Wa_b, const float* __restrict__ Ua_b,
    float* __restrict__ wa) {
  __shared__ float hs[H];
  int t = threadIdx.x;
  hs[t] = hid[t];
  __syncthreads();
  float s = 0.f;
  const float* row = Wa_w + t * H;
#pragma unroll 4
  for (int j = 0; j < H; ++j) s += row[j] * hs[j];
  wa[t] = s + Wa_b[t] + Ua_b[t];
}

// ---------------------------------------------------------------------------
// Fused attention double-GEMM per 16-row tile of the sequence:
//   T[m,k]    = tanh(wa[k] + sum_j enc[s0+m, j] * Ua_w[k, j])
//   Anxn[m,n] = Va_b[n] + sum_k T[m,k] * Va_w[n, k]
// Block = 512 threads = 16 waves; wave w owns N-tile w (columns w*16..w*16+15).
// The 16xH encoder tile is staged into LDS by the Tensor Data Mover (TDM),
// using descriptor padding (4 DWORDs per 256-DWORD row) to produce the
// bank-conflict-free LDSP=260 layout directly.
__global__ __launch_bounds__(512) void attn_kernel(
    const float* __restrict__ enc, const float* __restrict__ Ua_w,
    const float* __restrict__ Va_w, const float* __restrict__ Va_b,
    const float* __restrict__ wa, float* __restrict__ A_buf) {
  __shared__ float E[16][LDSP];
  __shared__ float T[16][LDSP];
  const int tid  = threadIdx.x;
  const int wave = tid >> 5;
  const int lane = tid & 31;
  const int s0   = blockIdx.x * 16;

#if __has_builtin(__builtin_amdgcn_tensor_load_to_lds)
  if (wave == 0) {
    // ---- Tensor DMA descriptor (D#) -------------------------------------
    unsigned long long gaddr =
        (unsigned long long)(uintptr_t)(enc + (size_t)s0 * H);
    unsigned lds_addr = (unsigned)(uintptr_t)&E[0][0];  // low 32b = LDS offset
    v4u g0;
    g0[0] = 1u;                                   // count=1, no gather
    g0[1] = lds_addr;                             // lds_addr[31:0]
    g0[2] = (unsigned)(gaddr & 0xFFFFFFFFu);      // global_addr[31:0]
    g0[3] = (unsigned)((gaddr >> 32) & 0x1FFFFFFu) | (2u << 30);  // type=2
    v8i g1;
    g1[0] = (int)((2u << 16)       // data_size = 4 bytes
                | (1u << 20)       // pad_enable
                | (7u << 22)       // pad_interval: every 256 DWORDs
                | (3u << 25));     // pad_amount: 4 DWORDs  -> 260-float rows
    g1[1] = (int)(256u << 16);     // tensor_dim0[15:0] = 256 (bits 79:64... low half)
    g1[2] = (int)(4096u << 16);    // tensor_dim0 hi = 0 | tensor_dim1 lo = 4096
    g1[3] = (int)(256u << 16);     // tensor_dim1 hi = 0 | tile_dim0 = 256
    g1[4] = 16;                    // tile_dim1 = 16 rows | tile_dim2 = 0
    g1[5] = 256;                   // tensor_dim0_stride = 256 elements
    g1[6] = 0;
    g1[7] = 0;
    v4i gz = {0, 0, 0, 0};
#if defined(__clang_major__) && (__clang_major__ >= 23)
    v8i gz8 = {0, 0, 0, 0, 0, 0, 0, 0};
    __builtin_amdgcn_tensor_load_to_lds(g0, g1, gz, gz, gz8, 0);
#else
    __builtin_amdgcn_tensor_load_to_lds(g0, g1, gz, gz, 0);
#endif
    __builtin_amdgcn_s_wait_tensorcnt(0);
  }
#else
  for (int i = tid; i < 16 * H; i += 512) {
    int r = i >> 8, c = i & 255;
    E[r][c] = enc[(s0 + r) * H + c];
  }
#endif
  __syncthreads();

  const int n0 = wave * 16;
  const int m  = lane & 15;          // A-frag row  (lanes 0-15 / 16-31 same M)
  const int n  = lane & 15;          // B/C-frag col
  const int hi = lane >> 4;          // 0: K0/K1 & M=r ; 1: K2/K3 & M=r+8
  const int kb = hi * 2;

  // -------- GEMM 1: enc x Ua_w^T ------------------------------------------
  v8f acc = {};
  for (int k0 = 0; k0 < H; k0 += 4) {
    if ((k0 & 63) == 0)  // software-prefetch phase-2 B panel (global_prefetch_b8)
      __builtin_prefetch(&Va_w[(n0 + n) * H + k0 + kb], 0, 1);
    v2f a, b;
    a.x = E[m][k0 + kb];
    a.y = E[m][k0 + kb + 1];
    b.x = Ua_w[(n0 + n) * H + k0 + kb];
    b.y = Ua_w[(n0 + n) * H + k0 + kb + 1];
    acc = __builtin_amdgcn_wmma_f32_16x16x4_f32(false, a, false, b,
                                                (short)0, acc, false, false);
  }
  const float addn = wa[n0 + n];
#pragma unroll
  for (int r = 0; r < 8; ++r) {
    int mm = r + (hi << 3);          // C layout: VGPR r -> M=r (lo) / M=r+8 (hi)
    T[mm][n0 + n] = tanhf(acc[r] + addn);
  }
  __syncthreads();

  // -------- GEMM 2: T x Va_w^T --------------------------------------------
  v8f acc2 = {};
  for (int k0 = 0; k0 < H; k0 += 4) {
    v2f a, b;
    a.x = T[m][k0 + kb];
    a.y = T[m][k0 + kb + 1];
    b.x = Va_w[(n0 + n) * H + k0 + kb];
    b.y = Va_w[(n0 + n) * H + k0 + kb + 1];
    acc2 = __builtin_amdgcn_wmma_f32_16x16x4_f32(false, a, false, b,
                                                 (short)0, acc2, false, false);
  }
  const float bias = Va_b[n0 + n];
#pragma unroll
  for (int r = 0; r < 8; ++r) {
    int mm = r + (hi << 3);
    A_buf[(size_t)(s0 + mm) * H + (n0 + n)] = acc2[r] + bias;
  }
}

// ---------------------------------------------------------------------------
// Stage 1 of column-wise logsumexp over s: block b reduces rows [b*64, b*64+64)
// with fully coalesced row reads; thread t owns column h=t (online update).
__global__ __launch_bounds__(H) void lse1_kernel(
    const float* __restrict__ A_buf, float* __restrict__ part_m,
    float* __restrict__ part_s) {
  int t = threadIdx.x;
  int sbeg = blockIdx.x * 64;
  float m = -INFINITY, sum = 0.f;
  for (int s = sbeg; s < sbeg + 64; ++s) {
    float v = A_buf[(size_t)s * H + t];
    if (v > m) { sum = sum * expf(m - v) + 1.f; m = v; }
    else       { sum += expf(v - m); }
  }
  part_m[blockIdx.x * H + t] = m;
  part_s[blockIdx.x * H + t] = sum;
}

// Stage 2: merge 64 partials per column; Ci[h] = (Anxn[0,h] - lse[h]) * enc[0,h]
__global__ __launch_bounds__(H) void lse2_kernel(
    const float* __restrict__ part_m, const float* __restrict__ part_s,
    const float* __restrict__ A_buf, const float* __restrict__ enc,
    float* __restrict__ Ci) {
  int h = threadIdx.x;
  float m = -INFINITY, sum = 0.f;
  for (int b = 0; b < 64; ++b) {
    float mb = part_m[b * H + h], sb = part_s[b * H + h];
    if (mb > m) { sum = sum * expf(m - mb) + sb; m = mb; }
    else        { sum += sb * expf(mb - m); }
  }
  float lse = m + logf(sum);
  Ci[h] = (A_buf[h] - lse) * enc[h];   // row s=0 only (reference quirk)
}

// ---------------------------------------------------------------------------
// u_buf[mat*H + h] = dot(U*_w[h, :], emb[id, :])  — 30.7 MB streamed, float4.
__global__ __launch_bounds__(H) void bigU_kernel(
    const int* __restrict__ ids, const float* __restrict__ emb,
    const float* __restrict__ Uz_w, const float* __restrict__ Ur_w,
    const float* __restrict__ Uh_w, float* __restrict__ u_buf) {
  int mat = blockIdx.x >> 8;
  int h   = blockIdx.x & 255;
  const float* W = (mat == 0) ? Uz_w : (mat == 1) ? Ur_w : Uh_w;
  int id = ids[0];
  const float4* x4 = (const float4*)(emb + (size_t)id * VOCAB);
  const float4* w4 = (const float4*)(W + (size_t)h * VOCAB);
  float s = 0.f;
  for (int i = threadIdx.x; i < VOCAB / 4; i += H) {
    float4 a = x4[i], b = w4[i];
    s += a.x * b.x + a.y * b.y + a.z * b.z + a.w * b.w;
  }
  __shared__ float red[H];
  red[threadIdx.x] = s;
  __syncthreads();
  for (int st = H / 2; st > 0; st >>= 1) {
    if (threadIdx.x < st) red[threadIdx.x] += red[threadIdx.x + st];
    __syncthreads();
  }
  if (threadIdx.x == 0) u_buf[mat * H + h] = red[0];
}

// ---------------------------------------------------------------------------
// GRU gates; one block, thread h owns output channel h.
__global__ __launch_bounds__(H) void gru_kernel(
    const float* __restrict__ hid, const float* __restrict__ u_buf,
    const float* __restrict__ Uz_b, const float* __restrict__ Ur_b,
    const float* __restrict__ Uh_b,
    const float* __restrict__ Wz_w, const float* __restrict__ Wz_b,
    const float* __restrict__ Wr_w, const float* __restrict__ Wr_b,
    const float* __restrict__ Wh_w, const float* __restrict__ Wh_b,
    const float* __restrict__ Cz_w, const float* __restrict__ Cz_b,
    const float* __restrict__ Cr_w, const float* __restrict__ Cr_b,
    const float* __restrict__ Ch_w, const float* __restrict__ Ch_b,
    const float* __restrict__ Ci, float* __restrict__ nh_ws,
    float* __restrict__ out_hidden) {
  __shared__ float hs[H], cs[H], rh[H];
  int t = threadIdx.x;
  hs[t] = hid[t];
  cs[t] = Ci[t];
  __syncthreads();
  float az = u_buf[t]     + Uz_b[t] + Wz_b[t] + Cz_b[t];
  float ar = u_buf[H + t] + Ur_b[t] + Wr_b[t] + Cr_b[t];
#pragma unroll 4
  for (int j = 0; j < H; ++j) {
    az += Wz_w[t * H + j] * hs[j] + Cz_w[t * H + j] * cs[j];
    ar += Wr_w[t * H + j] * hs[j] + Cr_w[t * H + j] * cs[j];
  }
  float z = 1.f / (1.f + expf(-az));
  float r = 1.f / (1.f + expf(-ar));
  rh[t] = r * hs[t];
  __syncthreads();
  float ac = u_buf[2 * H + t] + Uh_b[t] + Wh_b[t] + Ch_b[t];
#pragma unroll 4
  for (int j = 0; j < H; ++j)
    ac += Wh_w[t * H + j] * rh[j] + Ch_w[t * H + j] * cs[j];
  float c = tanhf(ac);
  float nh = (1.f - z) * c + z * hs[t];
  nh_ws[t] = nh;
  out_hidden[t] = nh;
}

// ---------------------------------------------------------------------------
// logits[v] = dot(V_w[v,:], nh) + V_b[v]; one wave per row, coalesced.
__global__ __launch_bounds__(H) void vproj_kernel(
    const float* __restrict__ V_w, const float* __restrict__ V_b,
    const float* __restrict__ nh_ws, float* __restrict__ logits) {
  __shared__ float nh[H];
  int t = threadIdx.x;
  nh[t] = nh_ws[t];
  __syncthreads();
  int row  = blockIdx.x * 8 + (t >> 5);
  int lane = t & 31;
  float s = 0.f;
  if (row < VOCAB) {
    const float* wrow = V_w + (size_t)row * H;
#pragma unroll
    for (int j = lane; j < H; j += 32) s += wrow[j] * nh[j];
  }
  for (int off = 16; off > 0; off >>= 1) s += __shfl_down(s, off, 32);
  if (lane == 0 && row < VOCAB) logits[row] = s + V_b[row];
}

// ---------------------------------------------------------------------------
// Final log_softmax over 10000 logits; single block.
__global__ __launch_bounds__(1024) void lsm_kernel(
    const float* __restrict__ logits, float* __restrict__ out) {
  __shared__ float red[1024];
  int t = threadIdx.x;
  float m = -INFINITY;
  for (int i = t; i < VOCAB; i += 1024) m = fmaxf(m, logits[i]);
  red[t] = m;
  __syncthreads();
  for (int st = 512; st > 0; st >>= 1) {
    if (t < st) red[t] = fmaxf(red[t], red[t + st]);
    __syncthreads();
  }
  float gm = red[0];
  __syncthreads();
  float s = 0.f;
  for (int i = t; i < VOCAB; i += 1024) s += expf(logits[i] - gm);
  red[t] = s;
  __syncthreads();
  for (int st = 512; st > 0; st >>= 1) {
    if (t < st) red[t] += red[t + st];
    __syncthreads();
  }
  float lse = gm + logf(red[0]);
  for (int i = t; i < VOCAB; i += 1024) out[i] = logits[i] - lse;
}

// ---------------------------------------------------------------------------
extern "C" void kernel_launch(void* const* d_in, const int* in_sizes, int n_in,
                              void* d_out, int out_size, void* d_ws, size_t ws_size,
                              hipStream_t stream) {
  // setup_inputs() order: input_ids, hidden, encoder_outputs, then params dict
  const int*   ids  = (const int*)  d_in[0];
  const float* hid  = (const float*)d_in[1];
  const float* enc  = (const float*)d_in[2];
  const float* Uz_w = (const float*)d_in[3],  *Uz_b = (const float*)d_in[4];
  const float* Ur_w = (const float*)d_in[5],  *Ur_b = (const float*)d_in[6];
  const float* Uh_w = (const float*)d_in[7],  *Uh_b = (const float*)d_in[8];
  const float* Wz_w = (const float*)d_in[9],  *Wz_b = (const float*)d_in[10];
  const float* Wr_w = (const float*)d_in[11], *Wr_b = (const float*)d_in[12];
  const float* Wh_w = (const float*)d_in[13], *Wh_b = (const float*)d_in[14];
  const float* Cz_w = (const float*)d_in[15], *Cz_b = (const float*)d_in[16];
  const float* Cr_w = (const float*)d_in[17], *Cr_b = (const float*)d_in[18];
  const float* Ch_w = (const float*)d_in[19], *Ch_b = (const float*)d_in[20];
  const float* Wa_w = (const float*)d_in[21], *Wa_b = (const float*)d_in[22];
  const float* Ua_w = (const float*)d_in[23], *Ua_b = (const float*)d_in[24];
  const float* Va_w = (const float*)d_in[25], *Va_b = (const float*)d_in[26];
  const float* V_w  = (const float*)d_in[27], *V_b  = (const float*)d_in[28];
  const float* emb  = (const float*)d_in[29];

  float* out = (float*)d_out;               // [0,10000) output ; [10000,10256) new_hidden

  // workspace layout (floats)
  float* ws     = (float*)d_ws;
  float* wa     = ws;                       // 256
  float* Ci     = ws + 256;                 // 256
  float* u_buf  = ws + 512;                 // 768
  float* nh     = ws + 1280;                // 256
  float* logits = ws + 1536;                // 10000
  float* part_m = ws + 11536;               // 64*256
  float* part_s = ws + 27920;               // 64*256
  float* A_buf  = ws + 44304;               // 4096*256  (total ~4.4 MB)

  prep_wa_kernel<<<1, H, 0, stream>>>(hid, Wa_w, Wa_b, Ua_b, wa);
  attn_kernel<<<S / 16, 512, 0, stream>>>(enc, Ua_w, Va_w, Va_b, wa, A_buf);
  lse1_kernel<<<S / 64, H, 0, stream>>>(A_buf, part_m, part_s);
  lse2_kernel<<<1, H, 0, stream>>>(part_m, part_s, A_buf, enc, Ci);
  bigU_kernel<<<3 * H, H, 0, stream>>>(ids, emb, Uz_w, Ur_w, Uh_w, u_buf);
  gru_kernel<<<1, H, 0, stream>>>(hid, u_buf, Uz_b, Ur_b, Uh_b,
                                  Wz_w, Wz_b, Wr_w, Wr_b, Wh_w, Wh_b,
                                  Cz_w, Cz_b, Cr_w, Cr_b, Ch_w, Ch_b,
                                  Ci, nh, out + VOCAB);
  vproj_kernel<<<VOCAB / 8, H, 0, stream>>>(V_w, V_b, nh, logits);
  lsm_kernel<<<1, 1024, 0, stream>>>(logits, out);
}